// GCN2Layer_7541962571801
// MI455X (gfx1250) — compile-verified
//
#include <hip/hip_runtime.h>
#include <hip/hip_bf16.h>
#include <math.h>

#define N_NODES 50000
#define N_EDGES 800000
#define IN_FEAT 512
#define HIDDEN  64
#define N_CLASS 16

typedef __attribute__((ext_vector_type(2))) float v2f;
typedef __attribute__((ext_vector_type(8))) float v8f;

// ---------------------------------------------------------------------------
// Repack W1[512,64] into WMMA B-fragment order:
//   B1p[((t*32 + lane)*4 + n)*2 + j] = W1[4t + 2*(lane>>4) + j][16n + (lane&15)]
// so each lane's per-k-step data for all 4 N-tiles is 32 contiguous bytes
// (2x global_load_b128 per k-step in the GEMM).
// ---------------------------------------------------------------------------
__global__ void gcn_pack_w1(const float* __restrict__ W1, float* __restrict__ B1p) {
    const int i = blockIdx.x * blockDim.x + threadIdx.x;
    if (i >= (IN_FEAT / 4) * 32 * 8) return;
    const int j    = i & 1;
    const int n    = (i >> 1) & 3;
    const int lane = (i >> 3) & 31;
    const int t    = i >> 8;
    const int k    = 4 * t + 2 * (lane >> 4) + j;
    const int col  = 16 * n + (lane & 15);
    B1p[i] = W1[(size_t)k * HIDDEN + col];
}

// Same idea for W2[64,16] (single N-tile): B2p[(t*32 + lane)*2 + j]
__global__ void gcn_pack_w2(const float* __restrict__ W2, float* __restrict__ B2p) {
    const int i = blockIdx.x * blockDim.x + threadIdx.x;
    if (i >= (HIDDEN / 4) * 32 * 2) return;
    const int j    = i & 1;
    const int lane = (i >> 1) & 31;
    const int t    = i >> 6;
    const int k    = 4 * t + 2 * (lane >> 4) + j;
    B2p[i] = W2[(size_t)k * N_CLASS + (lane & 15)];
}

// ---------------------------------------------------------------------------
// GEMM1: H0[N,64] = x[N,512] @ W1[512,64] via V_WMMA_F32_16X16X4_F32.
// One wave computes a full 16x64 output slab (4 accumulators), so each A
// fragment (and each HBM byte of x) is consumed by 4 WMMAs -> x read once.
// Inner loop per k-step: 1x b64 (A) + 2x b128 (packed B) + 4x WMMA.
// ---------------------------------------------------------------------------
__global__ void gcn_gemm1_wmma(const float* __restrict__ x,
                               const float* __restrict__ B1p,
                               float* __restrict__ H0) {
    const int tileM = blockIdx.x * 4 + threadIdx.y;   // wave-uniform
    if (tileM * 16 >= N_NODES) return;                // uniform exit, EXEC stays full
    const int lane = threadIdx.x;
    const int hi   = lane >> 4;
    const int l16  = lane & 15;

    const float* __restrict__ xrow  = x + (size_t)(tileM * 16 + l16) * IN_FEAT;
    const float* __restrict__ bbase = B1p + (size_t)lane * 8;

    v8f acc[4];
#pragma unroll
    for (int n = 0; n < 4; ++n) acc[n] = (v8f){0.f,0.f,0.f,0.f,0.f,0.f,0.f,0.f};

#pragma unroll 4
    for (int t = 0; t < IN_FEAT / 4; ++t) {
        // A 16x4 frag: lane(l16,hi) holds x[row=l16][4t+2hi .. 4t+2hi+1]
        const v2f a = *(const v2f*)(xrow + 4 * t + hi * 2);
        // Packed B: 32 contiguous bytes = fragments for all 4 N-tiles
        const float4* bp = (const float4*)(bbase + (size_t)t * 32 * 8);
        const float4 b01 = bp[0];
        const float4 b23 = bp[1];
        const v2f b[4] = { {b01.x, b01.y}, {b01.z, b01.w},
                           {b23.x, b23.y}, {b23.z, b23.w} };
#pragma unroll
        for (int n = 0; n < 4; ++n)
            acc[n] = __builtin_amdgcn_wmma_f32_16x16x4_f32(false, a, false, b[n],
                                                           (short)0, acc[n], false, false);
    }
    // C/D layout: VGPR v -> row (hi*8 + v), col l16 (+16n per tile)
    float* __restrict__ out = H0 + (size_t)(tileM * 16 + hi * 8) * HIDDEN + l16;
#pragma unroll
    for (int n = 0; n < 4; ++n)
#pragma unroll
        for (int v = 0; v < 8; ++v)
            out[(size_t)v * HIDDEN + n * 16] = acc[n][v];
}

// ---------------------------------------------------------------------------
// GEMM2: L[N,16] = relu(H1[N,64] + b1) @ W2[64,16]  (bias+ReLU fused into A)
// ---------------------------------------------------------------------------
__global__ void gcn_gemm2_wmma(const float* __restrict__ H1,
                               const float* __restrict__ b1,
                               const float* __restrict__ B2p,
                               float* __restrict__ L) {
    const int tileM = blockIdx.x * 4 + threadIdx.y;
    if (tileM * 16 >= N_NODES) return;
    const int lane = threadIdx.x;
    const int hi   = lane >> 4;
    const int l16  = lane & 15;

    const float* __restrict__ hrow = H1 + (size_t)(tileM * 16 + l16) * HIDDEN;

    v8f acc = {0.f, 0.f, 0.f, 0.f, 0.f, 0.f, 0.f, 0.f};

#pragma unroll
    for (int t = 0; t < HIDDEN / 4; ++t) {
        const int ka = 4 * t + hi * 2;
        v2f a;
        a.x = fmaxf(hrow[ka]     + b1[ka],     0.f);
        a.y = fmaxf(hrow[ka + 1] + b1[ka + 1], 0.f);
        const v2f b = *(const v2f*)(B2p + ((size_t)t * 32 + lane) * 2);
        acc = __builtin_amdgcn_wmma_f32_16x16x4_f32(false, a, false, b,
                                                    (short)0, acc, false, false);
    }
    float* __restrict__ out = L + (size_t)(tileM * 16 + hi * 8) * N_CLASS + l16;
#pragma unroll
    for (int v = 0; v < 8; ++v) out[(size_t)v * N_CLASS] = acc[v];
}

// ---------------------------------------------------------------------------
// Zero-fill (float4-vectorized)
// ---------------------------------------------------------------------------
__global__ void gcn_zero_f4(float4* __restrict__ p, int n4) {
    int i = blockIdx.x * blockDim.x + threadIdx.x;
    if (i < n4) p[i] = make_float4(0.f, 0.f, 0.f, 0.f);
}

// ---------------------------------------------------------------------------
// SpMM #1: H1[dst[e], :] += w[e] * H0[src[e], :]   (64 feats, float4 groups)
// 16 consecutive threads share one edge -> coalesced 256B row reads from the
// L2-resident H0 (12.8 MB << 192 MB L2); fp32 atomics resolve at L2.
// ---------------------------------------------------------------------------
__global__ void gcn_spmm1(const int* __restrict__ src, const int* __restrict__ dst,
                          const float* __restrict__ ew,
                          const float* __restrict__ H0, float* __restrict__ H1) {
    const long long tid = (long long)blockIdx.x * blockDim.x + threadIdx.x;
    if (tid >= (long long)N_EDGES * 16) return;
    const int e  = (int)(tid >> 4);
    const int fg = (int)(tid & 15);
    const int s = src[e], d = dst[e];
    const float w = ew[e];
    const float4 m = *(const float4*)(H0 + (size_t)s * HIDDEN + fg * 4);
    float* o = H1 + (size_t)d * HIDDEN + fg * 4;
    atomicAdd(o + 0, w * m.x);
    atomicAdd(o + 1, w * m.y);
    atomicAdd(o + 2, w * m.z);
    atomicAdd(o + 3, w * m.w);
}

// ---------------------------------------------------------------------------
// SpMM #2: out[dst[e], :] += w[e] * L[src[e], :]   (16 feats, float4 groups)
// ---------------------------------------------------------------------------
__global__ void gcn_spmm2(const int* __restrict__ src, const int* __restrict__ dst,
                          const float* __restrict__ ew,
                          const float* __restrict__ L, float* __restrict__ out) {
    const long long tid = (long long)blockIdx.x * blockDim.x + threadIdx.x;
    if (tid >= (long long)N_EDGES * 4) return;
    const int e  = (int)(tid >> 2);
    const int fg = (int)(tid & 3);
    const int s = src[e], d = dst[e];
    const float w = ew[e];
    const float4 m = *(const float4*)(L + (size_t)s * N_CLASS + fg * 4);
    float* o = out + (size_t)d * N_CLASS + fg * 4;
    atomicAdd(o + 0, w * m.x);
    atomicAdd(o + 1, w * m.y);
    atomicAdd(o + 2, w * m.z);
    atomicAdd(o + 3, w * m.w);
}

// ---------------------------------------------------------------------------
// In-place bias + softmax over 16 classes, one node per thread.
// ---------------------------------------------------------------------------
__global__ void gcn_bias_softmax(float* __restrict__ logits, const float* __restrict__ b2) {
    const int n = blockIdx.x * blockDim.x + threadIdx.x;
    if (n >= N_NODES) return;
    float4* row = (float4*)(logits + (size_t)n * N_CLASS);
    const float4* bias = (const float4*)b2;
    float v[N_CLASS];
#pragma unroll
    for (int g = 0; g < 4; ++g) {
        float4 r = row[g], b = bias[g];
        v[g * 4 + 0] = r.x + b.x; v[g * 4 + 1] = r.y + b.y;
        v[g * 4 + 2] = r.z + b.z; v[g * 4 + 3] = r.w + b.w;
    }
    float m = v[0];
#pragma unroll
    for (int i = 1; i < N_CLASS; ++i) m = fmaxf(m, v[i]);
    float s = 0.f;
#pragma unroll
    for (int i = 0; i < N_CLASS; ++i) { v[i] = __expf(v[i] - m); s += v[i]; }
    const float inv = 1.f / s;
#pragma unroll
    for (int g = 0; g < 4; ++g)
        row[g] = make_float4(v[g*4+0]*inv, v[g*4+1]*inv, v[g*4+2]*inv, v[g*4+3]*inv);
}

extern "C" void kernel_launch(void* const* d_in, const int* in_sizes, int n_in,
                              void* d_out, int out_size, void* d_ws, size_t ws_size,
                              hipStream_t stream) {
    const float* x   = (const float*)d_in[0];
    const int*   src = (const int*)  d_in[1];
    const int*   dst = (const int*)  d_in[2];
    const float* ew  = (const float*)d_in[3];
    const float* W1  = (const float*)d_in[4];
    const float* b1  = (const float*)d_in[5];
    const float* W2  = (const float*)d_in[6];
    const float* b2  = (const float*)d_in[7];
    float* out = (float*)d_out;

    float* H0  = (float*)d_ws;                      // [N,64]  12.8 MB
    float* H1  = H0 + (size_t)N_NODES * HIDDEN;     // [N,64]  12.8 MB
    float* L   = H1 + (size_t)N_NODES * HIDDEN;     // [N,16]   3.2 MB
    float* B1p = L  + (size_t)N_NODES * N_CLASS;    // 32768 floats, 128 KB
    float* B2p = B1p + (IN_FEAT / 4) * 32 * 8;      // 1024 floats, 4 KB

    const int rowTiles = N_NODES / 16;              // 3125 (exact)
    const int gemmGx   = (rowTiles + 3) / 4;        // 782

    // 0) Pack W1/W2 into WMMA B-fragment order (L2-resident, coalesced loads)
    gcn_pack_w1<<<((IN_FEAT / 4) * 32 * 8 + 255) / 256, 256, 0, stream>>>(W1, B1p);
    gcn_pack_w2<<<((HIDDEN / 4) * 32 * 2 + 255) / 256, 256, 0, stream>>>(W2, B2p);

    // 1) H0 = x @ W1   (x streamed from HBM exactly once)
    gcn_gemm1_wmma<<<dim3(gemmGx), dim3(32, 4), 0, stream>>>(x, B1p, H0);

    // 2) H1 = 0 ; H1 += adj-scatter of H0
    {
        int n4 = N_NODES * HIDDEN / 4;
        gcn_zero_f4<<<(n4 + 255) / 256, 256, 0, stream>>>((float4*)H1, n4);
    }
    {
        long long t = (long long)N_EDGES * 16;
        gcn_spmm1<<<(unsigned)((t + 255) / 256), 256, 0, stream>>>(src, dst, ew, H0, H1);
    }

    // 3) L = relu(H1 + b1) @ W2
    gcn_gemm2_wmma<<<dim3(gemmGx), dim3(32, 4), 0, stream>>>(H1, b1, B2p, L);

    // 4) out = 0 ; out += adj-scatter of L
    {
        int n4 = N_NODES * N_CLASS / 4;
        gcn_zero_f4<<<(n4 + 255) / 256, 256, 0, stream>>>((float4*)out, n4);
    }
    {
        long long t = (long long)N_EDGES * 4;
        gcn_spmm2<<<(unsigned)((t + 255) / 256), 256, 0, stream>>>(src, dst, ew, L, out);
    }

    // 5) out = softmax(out + b2)
    gcn_bias_softmax<<<(N_NODES + 255) / 256, 256, 0, stream>>>(out, b2);
}